// Sampler_53317724012662
// MI455X (gfx1250) — compile-verified
//
#include <hip/hip_runtime.h>
#include <stdint.h>

#define TPB    256
#define NWAVES 8
#define CAND   2048
#define HPAD   257   // 257 dwords stride -> per-wave hist copies in different banks

__device__ __forceinline__ uint32_t orderKey(float f) {
    uint32_t u = __float_as_uint(f);
    return (u & 0x80000000u) ? ~u : (u | 0x80000000u);
}
__device__ __forceinline__ float keyToFloat(uint32_t k) {
    uint32_t u = (k & 0x80000000u) ? (k & 0x7FFFFFFFu) : ~k;
    return __uint_as_float(u);
}

__global__ __launch_bounds__(TPB) void sampler_kernel(
    const float* __restrict__ logits,
    const float* __restrict__ temps,
    const int*   __restrict__ topks,
    const float* __restrict__ topps,
    const float* __restrict__ uvals,
    int* __restrict__ out, int V)
{
    __shared__ float4 stage[NWAVES][2][32];          // 8 KB async staging
    __shared__ uint32_t hist[NWAVES][HPAD];          // privatized radix histograms
    __shared__ float redM[NWAVES], redS[NWAVES];
    __shared__ float shM, shS;
    __shared__ uint32_t shPrefix, shKrem;
    __shared__ unsigned long long cand[CAND];        // 16 KB candidates
    __shared__ uint32_t candCnt;

    const int row  = blockIdx.x;
    const int tid  = threadIdx.x;
    const int lane = tid & 31;
    const int wave = tid >> 5;

    const float T  = temps[row];
    const float Tc = fmaxf(T, 1e-10f);
    const float* rowp = logits + (size_t)row * (size_t)V;
    const float4* rowp4 = (const float4*)rowp;
    const uint64_t base = (uint64_t)(uintptr_t)rowp;

    const int niter = (V + 1023) >> 10;              // 1024 floats / block-iter
    uint32_t ldsAddr[2];
    ldsAddr[0] = (uint32_t)(uintptr_t)&stage[wave][0][lane];
    ldsAddr[1] = (uint32_t)(uintptr_t)&stage[wave][1][lane];

    // ---------------- Pass 1: async HBM->LDS stream + online softmax --------
    float m = -INFINITY, s = 0.0f;
    {
        int e0 = wave * 128 + lane * 4;
        if (0 < niter && e0 + 4 <= V) {
            uint32_t goff = (uint32_t)e0 * 4u;
            asm volatile("global_load_async_to_lds_b128 %0, %1, %2"
                         :: "v"(ldsAddr[0]), "v"(goff), "s"(base) : "memory");
        }
        int e1 = 1024 + wave * 128 + lane * 4;
        if (1 < niter && e1 + 4 <= V) {
            uint32_t goff = (uint32_t)e1 * 4u;
            asm volatile("global_load_async_to_lds_b128 %0, %1, %2"
                         :: "v"(ldsAddr[1]), "v"(goff), "s"(base) : "memory");
        }
    }
    for (int t = 0; t < niter; ++t) {
        if (t == niter - 1)
            asm volatile("s_wait_asynccnt 0x0" ::: "memory");
        else
            asm volatile("s_wait_asynccnt 0x1" ::: "memory");
        int e = t * 1024 + wave * 128 + lane * 4;
        if (e + 4 <= V) {
            float4 v = stage[wave][t & 1][lane];
            float xs[4] = {v.x, v.y, v.z, v.w};
            #pragma unroll
            for (int j = 0; j < 4; ++j) {
                float x = xs[j] / Tc;
                if (x > m) { s = s * expf(m - x) + 1.0f; m = x; }
                else       { s += expf(x - m); }
            }
        }
        asm volatile("s_wait_dscnt 0x0" ::: "memory");  // LDS reads done before reuse
        int tn = t + 2;
        if (tn < niter) {
            int en = tn * 1024 + wave * 128 + lane * 4;
            if (en + 4 <= V) {
                uint32_t goff = (uint32_t)en * 4u;
                asm volatile("global_load_async_to_lds_b128 %0, %1, %2"
                             :: "v"(ldsAddr[t & 1]), "v"(goff), "s"(base) : "memory");
            }
        }
    }
    // wave32 reduction of (m, s)
    #pragma unroll
    for (int off = 16; off > 0; off >>= 1) {
        float om = __shfl_xor(m, off, 32);
        float os = __shfl_xor(s, off, 32);
        float nm = fmaxf(m, om);
        s = s * expf(m - nm) + os * expf(om - nm);
        m = nm;
    }
    if (lane == 0) { redM[wave] = m; redS[wave] = s; }
    __syncthreads();
    if (tid == 0) {
        float mm = redM[0], ss = redS[0];
        for (int w = 1; w < NWAVES; ++w) {
            float om = redM[w], os = redS[w];
            float nm = fmaxf(mm, om);
            ss = ss * expf(mm - nm) + os * expf(om - nm);
            mm = nm;
        }
        shM = mm; shS = ss;
        shPrefix = 0u;
        int kk = topks[row];
        long long effk = (kk == -1) ? (long long)V : (long long)kk;
        long long K = effk < 1 ? 1 : (effk > 1024 ? 1024 : effk);
        shKrem = (uint32_t)K;
        candCnt = 0u;
    }
    __syncthreads();
    const float fm = shM, fS = shS;
    const int V4 = V >> 2;

    // ---------------- 4-round radix select on prob key (L2-resident) --------
    for (int r = 0; r < 4; ++r) {
        const int shift = 24 - 8 * r;
        const uint32_t pmask = (r == 0) ? 0u : (0xFFFFFFFFu << (shift + 8));
        const uint32_t pref  = shPrefix;
        const uint32_t Krem  = shKrem;
        for (int w = 0; w < NWAVES; ++w)
            for (int i = tid; i < 256; i += TPB) hist[w][i] = 0u;
        __syncthreads();
        for (int i = tid; i < V4; i += TPB) {
            float4 f = rowp4[i];
            float xs[4] = {f.x, f.y, f.z, f.w};
            #pragma unroll
            for (int j = 0; j < 4; ++j) {
                float p = expf(xs[j] / Tc - fm) / fS;
                uint32_t key = orderKey(p);
                if ((key & pmask) == pref)
                    atomicAdd(&hist[wave][(key >> shift) & 255u], 1u);
            }
        }
        for (int i = (V4 << 2) + tid; i < V; i += TPB) {
            float p = expf(rowp[i] / Tc - fm) / fS;
            uint32_t key = orderKey(p);
            if ((key & pmask) == pref)
                atomicAdd(&hist[wave][(key >> shift) & 255u], 1u);
        }
        __syncthreads();
        if (tid < 256) {                    // fold privatized copies into hist[0]
            uint32_t t0 = hist[0][tid];
            #pragma unroll
            for (int w = 1; w < NWAVES; ++w) t0 += hist[w][tid];
            hist[0][tid] = t0;
        }
        __syncthreads();
        if (tid == 0) {
            uint32_t c = 0, D = 0, newKrem = Krem;
            for (int d = 255; d >= 0; --d) {
                uint32_t h = hist[0][d];
                if (c + h >= Krem) { D = (uint32_t)d; newKrem = Krem - c; break; }
                c += h;
            }
            shPrefix = pref | (D << shift);
            shKrem = newKrem;
        }
        __syncthreads();
    }
    const uint32_t Tkey = shPrefix;

    // ---------------- Gather candidates (>= threshold, positive prob) -------
    for (int i = tid; i < V4; i += TPB) {
        float4 f = rowp4[i];
        float xs[4] = {f.x, f.y, f.z, f.w};
        #pragma unroll
        for (int j = 0; j < 4; ++j) {
            float p = expf(xs[j] / Tc - fm) / fS;
            uint32_t key = orderKey(p);
            if (key >= Tkey && key > 0x80000000u) {       // exclude prob == 0
                uint32_t idx = (uint32_t)(i * 4 + j);
                uint32_t slot = atomicAdd(&candCnt, 1u);
                if (slot < CAND)
                    cand[slot] = ((unsigned long long)key << 32) |
                                 (unsigned long long)(~idx);
            }
        }
    }
    for (int i = (V4 << 2) + tid; i < V; i += TPB) {
        float p = expf(rowp[i] / Tc - fm) / fS;
        uint32_t key = orderKey(p);
        if (key >= Tkey && key > 0x80000000u) {
            uint32_t slot = atomicAdd(&candCnt, 1u);
            if (slot < CAND)
                cand[slot] = ((unsigned long long)key << 32) |
                             (unsigned long long)(~(uint32_t)i);
        }
    }
    __syncthreads();
    const uint32_t cnt = candCnt < CAND ? candCnt : CAND;
    for (int i = tid; i < CAND; i += TPB)
        if ((uint32_t)i >= cnt) cand[i] = 0ull;
    __syncthreads();

    // ---------------- Bitonic sort descending (key desc, idx asc) -----------
    for (uint32_t k2 = 2; k2 <= CAND; k2 <<= 1) {
        for (uint32_t j = k2 >> 1; j > 0; j >>= 1) {
            for (uint32_t i = tid; i < CAND; i += TPB) {
                uint32_t ixj = i ^ j;
                if (ixj > i) {
                    unsigned long long a = cand[i], b = cand[ixj];
                    bool descBlk = ((i & k2) == 0);
                    if (descBlk ? (a < b) : (a > b)) { cand[i] = b; cand[ixj] = a; }
                }
            }
            __syncthreads();
        }
    }

    // ---------------- Serial decision (replicates reference formulas) -------
    if (tid == 0) {
        int token;
        if (T == 0.0f) {
            token = (int)(~(uint32_t)(cand[0] & 0xFFFFFFFFull));  // greedy
        } else {
            int kk = topks[row];
            long long effk = (kk == -1) ? (long long)V : (long long)kk;
            long long K = effk < 1 ? 1 : (effk > 1024 ? 1024 : effk);
            int n = (int)(K < (long long)cnt ? K : (long long)cnt);
            if (n < 1) n = 1;
            const float top_p = topps[row];
            const float u = uvals[row];
            float cum = 0.0f;
            int L = 0;
            for (int i = 0; i < n; ++i) {
                uint32_t key = (uint32_t)(cand[i] >> 32);
                float p = keyToFloat(key);
                float cumIncl = cum + p;
                // reference: (cdf - p) <= top_p, rank < eff_k, index 0 forced
                bool keep = (i == 0) ||
                            (((cumIncl - p) <= top_p) && ((long long)i < effk));
                if (!keep) break;
                cum = cumIncl;
                ++L;
            }
            float denom = fmaxf(cum, 1e-10f);
            int sel = -1;
            float c2 = 0.0f;
            for (int i = 0; i < L; ++i) {
                uint32_t key = (uint32_t)(cand[i] >> 32);
                float p = keyToFloat(key);
                c2 += p / denom;
                if (c2 >= u) { sel = i; break; }
            }
            if (sel < 0) sel = 0;                       // jnp.argmax all-False -> 0
            token = (int)(~(uint32_t)(cand[sel] & 0xFFFFFFFFull));
        }
        out[row] = token;
    }
}

extern "C" void kernel_launch(void* const* d_in, const int* in_sizes, int n_in,
                              void* d_out, int out_size, void* d_ws, size_t ws_size,
                              hipStream_t stream) {
    const float* logits = (const float*)d_in[0];
    const float* temps  = (const float*)d_in[1];
    const int*   topks  = (const int*)d_in[2];
    const float* topps  = (const float*)d_in[3];
    const float* u      = (const float*)d_in[4];
    const int N = in_sizes[1];
    const int V = in_sizes[0] / N;
    int* out = (int*)d_out;
    sampler_kernel<<<N, TPB, 0, stream>>>(logits, temps, topks, topps, u, out, V);
}